// GHMBCE_13503377179036
// MI455X (gfx1250) — compile-verified
//
#include <hip/hip_runtime.h>
#include <hip/hip_bf16.h>
#include <math.h>

typedef __attribute__((ext_vector_type(2))) float v2f;
typedef __attribute__((ext_vector_type(8))) float v8f;

#define GHM_N     16384
#define GHM_DELTA 0.1f
#define GHM_EPS   1e-12f

__device__ __forceinline__ float softplus_f(float x) {
    // numerically stable softplus: max(x,0) + log1p(exp(-|x|))
    return fmaxf(x, 0.0f) + log1pf(expf(-fabsf(x)));
}

// ---------------- Kernel A: per-element g and BCE term ----------------
__global__ void ghm_elem_kernel(const float* __restrict__ logits,
                                const int*   __restrict__ targets,
                                const float* __restrict__ pos_weight,
                                float* __restrict__ g_out,
                                float* __restrict__ pe_out, int n) {
    int i = blockIdx.x * blockDim.x + threadIdx.x;
    if (i >= n) return;
    float x = logits[i];
    float y = (float)targets[i];
    float pred = 1.0f / (1.0f + expf(-x));
    g_out[i] = fabsf(pred - y);
    float pw = pos_weight[0];
    // pw * y * softplus(-x) + (1-y) * softplus(x)
    pe_out[i] = pw * y * softplus_f(-x) + (1.0f - y) * softplus_f(x);
}

// ---------------- Kernel B: O(N^2) density counts via WMMA Gram tiles ----------------
// |gi - gj| <= delta  <=>  0.5*gi^2 + 0.5*gj^2 - 0.5*delta^2 <= gi*gj
// The 16x16 tile of gi*gj is one rank-1 WMMA: D = A(16x4,K=1 used) x B(4x16).
__global__ void ghm_count_kernel(const float* __restrict__ g,
                                 float* __restrict__ gd, int n) {
    __shared__ float ldsG[GHM_N];                 // 64 KB: entire g array LDS-resident
    for (int idx = threadIdx.x; idx < n; idx += blockDim.x)
        ldsG[idx] = g[idx];
    __syncthreads();

    const int lane = threadIdx.x & 31;
    const int wave = threadIdx.x >> 5;
    const int hi   = lane >> 4;                   // 0: lanes 0-15, 1: lanes 16-31
    const int lo   = lane & 15;
    const int i0   = (blockIdx.x * (blockDim.x >> 5) + wave) * 16;

    // A operand (16x4 f32): K=0 column = g_i in lanes 0-15 of VGPR0, rest zero
    v2f a;
    a.x = (hi == 0) ? ldsG[i0 + lo] : 0.0f;
    a.y = 0.0f;

    const float d2h = 0.5f * GHM_DELTA * GHM_DELTA;
    // Hoisted per-lane row constants; this lane's D rows are M = r + 8*hi
    float aHalf[8];
#pragma unroll
    for (int r = 0; r < 8; ++r) {
        float gi = ldsG[i0 + r + 8 * hi];
        aHalf[r] = 0.5f * gi * gi - d2h;
    }

    float cnt[8];
#pragma unroll
    for (int r = 0; r < 8; ++r) cnt[r] = 0.0f;

    for (int j0 = 0; j0 < n; j0 += 16) {
        float bv = ldsG[j0 + lo];                 // g_j for this lane's N column
        // B operand (4x16 f32): row K=0 = g_j in lanes 0-15 of VGPR0, rest zero
        v2f b;
        b.x = (hi == 0) ? bv : 0.0f;
        b.y = 0.0f;
        v8f c = {};
        c = __builtin_amdgcn_wmma_f32_16x16x4_f32(false, a, false, b,
                                                  (short)0, c, false, false);
        float bHalf = 0.5f * bv * bv;
#pragma unroll
        for (int r = 0; r < 8; ++r) {
            float t = aHalf[r] + bHalf;
            cnt[r] += (t <= c[r]) ? 1.0f : 0.0f; // counts < 2^24, exact in f32
        }
    }

    // Row-sum across the 16 lanes of each half-wave (xor masks stay in-half)
#pragma unroll
    for (int r = 0; r < 8; ++r) {
#pragma unroll
        for (int off = 1; off < 16; off <<= 1)
            cnt[r] += __shfl_xor(cnt[r], off, 32);
    }
#pragma unroll
    for (int r = 0; r < 8; ++r) {
        if (lo == r) gd[i0 + r + 8 * hi] = cnt[r];   // raw count (no /delta yet)
    }
}

// ---------------- Kernel C: final two-scalar reduction ----------------
__global__ void ghm_final_kernel(const float* __restrict__ gdcnt,
                                 const float* __restrict__ pe,
                                 float* __restrict__ out, int n) {
    __shared__ float s0[256];
    __shared__ float s1[256];
    float accW = 0.0f, accP = 0.0f;
    for (int i = threadIdx.x; i < n; i += blockDim.x) {
        float GD   = gdcnt[i] * (1.0f / GHM_DELTA);
        float beta = (float)n / (GD + GHM_EPS);
        float p    = pe[i];
        accW += beta * p;
        accP += p;
    }
    s0[threadIdx.x] = accW;
    s1[threadIdx.x] = accP;
    __syncthreads();
    for (int s = 128; s > 0; s >>= 1) {
        if (threadIdx.x < (unsigned)s) {
            s0[threadIdx.x] += s0[threadIdx.x + s];
            s1[threadIdx.x] += s1[threadIdx.x + s];
        }
        __syncthreads();
    }
    if (threadIdx.x == 0) {
        out[0] = s0[0] / (float)n;   // weighted mean
        out[1] = s1[0] / (float)n;   // pure mean
    }
}

extern "C" void kernel_launch(void* const* d_in, const int* in_sizes, int n_in,
                              void* d_out, int out_size, void* d_ws, size_t ws_size,
                              hipStream_t stream) {
    const float* logits  = (const float*)d_in[0];
    const int*   targets = (const int*)d_in[1];
    const float* pw      = (const float*)d_in[2];
    int n = in_sizes[0];                  // 16384

    float* g   = (float*)d_ws;            // [n]
    float* pe  = g + n;                    // [n]
    float* gdc = pe + n;                   // [n]  (192 KB total scratch)
    float* out = (float*)d_out;            // 2 floats

    ghm_elem_kernel<<<(n + 255) / 256, 256, 0, stream>>>(logits, targets, pw, g, pe, n);
    // 256 threads = 8 waves/block, one 16-row i-tile per wave -> 128 rows/block
    ghm_count_kernel<<<n / 128, 256, 0, stream>>>(g, gdc, n);
    ghm_final_kernel<<<1, 256, 0, stream>>>(gdc, pe, out, n);
}